// MS_SSIM_L1_LOSS_61753039782363
// MI455X (gfx1250) — compile-verified
//
#include <hip/hip_runtime.h>

// ---------------------------------------------------------------------------
// MS-SSIM + L1 loss, fully fused for MI455X (gfx1250, wave32, WMMA).
// Separable 33-tap Gaussian convs expressed as banded GEMMs on
// V_WMMA_F32_16X16X4_F32. One workgroup = one (batch, 16x16 tile).
// Round 4: async global->LDS staging with the correct builtin pointer types
// (int32x4 in AS1/AS3, per the compiler's own diagnostic); signal stride 52
// keeps LDS rows 16B-aligned for B128 async writes and bank-conflict-free.
// ---------------------------------------------------------------------------

typedef float v2f __attribute__((ext_vector_type(2)));
typedef float v8f __attribute__((ext_vector_type(8)));

#define BN     8
#define CN     3
#define HWD    512
#define TS     16            // output tile edge
#define HL     16            // halo (PAD) each side
#define HT     48            // TS + 2*HL
#define SW     52            // padded signal stride: 16B-aligned rows, 16 banks
#define PW     (TS + 1)      // padded hp/out stride
#define NCONV  38            // 7 unique (ch,sigma) combos * 5 moments + 3 L1
#define C1C    (0.01f * 0.01f)
#define C2C    (0.03f * 0.03f)

#if __has_builtin(__builtin_amdgcn_global_load_async_to_lds_b128) && \
    __has_builtin(__builtin_amdgcn_s_wait_asynccnt)
#define USE_ASYNC_LDS 1
typedef int v4i __attribute__((vector_size(4 * sizeof(int))));
typedef v4i __attribute__((address_space(1)))* GP128;   // global int32x4*
typedef v4i __attribute__((address_space(3)))* LP128;   // LDS    int32x4*
#endif

// --- prep: recover 1-D Gaussian factors from the 2-D g_base -----------------
// g2d = outer(g1,g1), sum(g1)=1  =>  g1[j] = g2d[16][j] / sum_j g2d[16][j]
__global__ void extract_g1(const float* __restrict__ gb, float* __restrict__ f1d) {
    int s = threadIdx.x;
    if (s < 5) {
        const float* row = gb + (size_t)s * 33 * 33 + 16 * 33;
        float sum = 0.f;
        for (int j = 0; j < 33; ++j) sum += row[j];
        float inv = 1.f / sum;
        for (int j = 0; j < 33; ++j) f1d[s * 33 + j] = row[j] * inv;
    }
}

// --- one separable conv, signal selected at compile time --------------------
// hpass: hp(48x16) = SIG(48x48) x Wt(48x16), Wt[k,j]=g1[k-j]
// vpass: out(16x16) = W(16x48) x hp(48x16),  W[m,k]=g1[k-m]
// Band fragments are the same values for hpass-B and vpass-A -> registers.
template <int SID>
__device__ __forceinline__ void conv_one(const float* __restrict__ sxc,
                                         const float* __restrict__ syc,
                                         const float* __restrict__ fr,
                                         float* __restrict__ hpw,
                                         float* __restrict__ dst,
                                         int l15, int half) {
    v2f w[12];
#pragma unroll
    for (int ks = 0; ks < 12; ++ks) {
        int k0 = ks * 4 + half * 2;              // f32 A/B: lane-half owns K pair
        w[ks].x = fr[k0     - l15 + 16];
        w[ks].y = fr[k0 + 1 - l15 + 16];
    }
#pragma unroll
    for (int rb = 0; rb < 3; ++rb) {
        int i0 = rb * 16;
        v8f acc = {};
#pragma unroll
        for (int ks = 0; ks < 12; ++ks) {
            int k0 = ks * 4 + half * 2;
            int base = (i0 + l15) * SW + k0;
            v2f a;
            if constexpr (SID == 0) {
                a.x = sxc[base]; a.y = sxc[base + 1];
            } else if constexpr (SID == 1) {
                a.x = syc[base]; a.y = syc[base + 1];
            } else if constexpr (SID == 2) {
                float u0 = sxc[base], u1 = sxc[base + 1];
                a.x = u0 * u0; a.y = u1 * u1;
            } else if constexpr (SID == 3) {
                float u0 = syc[base], u1 = syc[base + 1];
                a.x = u0 * u0; a.y = u1 * u1;
            } else if constexpr (SID == 4) {
                a.x = sxc[base] * syc[base];
                a.y = sxc[base + 1] * syc[base + 1];
            } else {
                a.x = fabsf(sxc[base] - syc[base]);
                a.y = fabsf(sxc[base + 1] - syc[base + 1]);
            }
            acc = __builtin_amdgcn_wmma_f32_16x16x4_f32(
                    false, a, false, w[ks], (short)0, acc, false, false);
        }
#pragma unroll
        for (int r8 = 0; r8 < 8; ++r8)           // C/D: lane-half selects M or M+8
            hpw[(i0 + r8 + half * 8) * PW + l15] = acc[r8];
    }
    v8f acc = {};
#pragma unroll
    for (int ks = 0; ks < 12; ++ks) {
        int k0 = ks * 4 + half * 2;
        v2f bv;
        bv.x = hpw[k0 * PW + l15];
        bv.y = hpw[(k0 + 1) * PW + l15];
        acc = __builtin_amdgcn_wmma_f32_16x16x4_f32(
                false, w[ks], false, bv, (short)0, acc, false, false);
    }
#pragma unroll
    for (int r8 = 0; r8 < 8; ++r8)
        dst[(r8 + half * 8) * PW + l15] = acc[r8];
}

// --- main fused kernel ------------------------------------------------------
__global__ __launch_bounds__(256)
void msssim_l1_fused(const float* __restrict__ x, const float* __restrict__ y,
                     const float* __restrict__ f1d, float* __restrict__ out) {
    __shared__ __align__(16) float sx[CN][HT][SW];   // x halo tiles
    __shared__ __align__(16) float sy[CN][HT][SW];   // y halo tiles
    __shared__ float hp[8][HT][PW];           // per-wave hpass scratch (48x16)
    __shared__ float ov[NCONV][TS][PW];       // conv outputs for combine
    __shared__ float fg[5][80];               // zero-padded 1-D filters
    __shared__ float red[256];

    const int tid  = threadIdx.x;
    const int lane = tid & 31;
    const int wave = tid >> 5;
    const int half = lane >> 4;               // 0: lanes 0-15, 1: lanes 16-31
    const int l15  = lane & 15;

    const int b    = blockIdx.x >> 10;        // 1024 tiles per image
    const int tile = blockIdx.x & 1023;
    const int ty   = tile >> 5, tx = tile & 31;
    const int row0 = ty * TS - HL;
    const int col0 = tx * TS - HL;

    // filters into padded LDS: fg[s][i] = g1[i-16] for i in [16,48], else 0.
    for (int i = tid; i < 5 * 80; i += 256) {
        int s = i / 80, j = i % 80;
        float v = 0.f;
        if (j >= 16 && j < 49) v = f1d[s * 33 + (j - 16)];
        fg[s][j] = v;
    }

    // halo tiles of x and y; interior tiles stage 16B chunks.
    const bool interior = (row0 >= 0) && (row0 + HT <= HWD) &&
                          (col0 >= 0) && (col0 + HT <= HWD);
    if (interior) {
        // col0 = 16*tx - 16 >= 0 is a multiple of 16 -> 16B aligned b128 transfers
        for (int i = tid; i < CN * HT * (HT / 4); i += 256) {
            int c  = i / (HT * (HT / 4));
            int rr = (i / (HT / 4)) % HT;
            int c4 = (i % (HT / 4)) * 4;
            size_t idx = ((size_t)(b * CN + c) * HWD + (row0 + rr)) * HWD + col0 + c4;
#ifdef USE_ASYNC_LDS
            // direct cache->LDS copy, tracked by ASYNCcnt (no VGPR round-trip)
            __builtin_amdgcn_global_load_async_to_lds_b128(
                (GP128)(x + idx), (LP128)(&sx[c][rr][c4]), 0, 0);
            __builtin_amdgcn_global_load_async_to_lds_b128(
                (GP128)(y + idx), (LP128)(&sy[c][rr][c4]), 0, 0);
#else
            float4 vx = *reinterpret_cast<const float4*>(x + idx);
            float4 vy = *reinterpret_cast<const float4*>(y + idx);
            float* dx = &sx[c][rr][c4];
            float* dy = &sy[c][rr][c4];
            dx[0] = vx.x; dx[1] = vx.y; dx[2] = vx.z; dx[3] = vx.w;
            dy[0] = vy.x; dy[1] = vy.y; dy[2] = vy.z; dy[3] = vy.w;
#endif
        }
#ifdef USE_ASYNC_LDS
        __builtin_amdgcn_s_wait_asynccnt(0);
#endif
    } else {
        for (int i = tid; i < CN * HT * HT; i += 256) {
            int c  = i / (HT * HT);
            int rr = (i / HT) % HT;
            int cc = i % HT;
            int gr = row0 + rr, gc = col0 + cc;
            float vx = 0.f, vy = 0.f;
            if (gr >= 0 && gr < HWD && gc >= 0 && gc < HWD) {
                size_t idx = ((size_t)(b * CN + c) * HWD + gr) * HWD + gc;
                vx = x[idx];
                vy = y[idx];
            }
            sx[c][rr][cc] = vx;
            sy[c][rr][cc] = vy;
        }
    }
    __syncthreads();

    // unique (channel, sigma-index) combos present in the 15 output channels
    const int combo_ch[7] = {0, 0, 1, 1, 1, 2, 2};
    const int combo_sg[7] = {0, 1, 1, 2, 3, 3, 4};

    // 38 separable convs distributed over 8 waves
    for (int cv = wave; cv < NCONV; cv += 8) {
        int ch, sg, sid;
        if (cv < 35) { int u = cv / 5; ch = combo_ch[u]; sg = combo_sg[u]; sid = cv % 5; }
        else         { ch = cv - 35;   sg = 4;           sid = 5; }
        const float* fr  = fg[sg];
        const float* sxc = &sx[ch][0][0];
        const float* syc = &sy[ch][0][0];
        float* hpw = &hp[wave][0][0];
        float* dst = &ov[cv][0][0];
        switch (sid) {                         // wave-uniform: one branch per conv
            case 0:  conv_one<0>(sxc, syc, fr, hpw, dst, l15, half); break;
            case 1:  conv_one<1>(sxc, syc, fr, hpw, dst, l15, half); break;
            case 2:  conv_one<2>(sxc, syc, fr, hpw, dst, l15, half); break;
            case 3:  conv_one<3>(sxc, syc, fr, hpw, dst, l15, half); break;
            case 4:  conv_one<4>(sxc, syc, fr, hpw, dst, l15, half); break;
            default: conv_one<5>(sxc, syc, fr, hpw, dst, l15, half); break;
        }
    }
    __syncthreads();

    // per-pixel SSIM combine; duplicate channels folded via multiplicities
    const int mult[7] = {3, 2, 1, 3, 1, 2, 3};       // sums to 15
    int pr = tid >> 4, pn = tid & 15;
    float PIcs = 1.f, l6 = 0.f;
#pragma unroll
    for (int u = 0; u < 7; ++u) {
        float mux = ov[u * 5 + 0][pr][pn];
        float muy = ov[u * 5 + 1][pr][pn];
        float mxx = ov[u * 5 + 2][pr][pn];
        float myy = ov[u * 5 + 3][pr][pn];
        float mxy = ov[u * 5 + 4][pr][pn];
        float mux2 = mux * mux, muy2 = muy * muy;
        float s2x = mxx - mux2, s2y = myy - muy2, sxy = mxy - mux * muy;
        float l  = (2.f * mux * muy + C1C) / (mux2 + muy2 + C1C);
        float cs = (2.f * sxy + C2C) / (s2x + s2y + C2C);
        float csm = cs;
        for (int m = 1; m < mult[u]; ++m) csm *= cs;
        PIcs *= csm;
        if (u == 6) l6 = l;                          // (ch2, sigma=8) -> lM = l^3
    }
    float lM  = l6 * l6 * l6;
    float ms  = 1.f - lM * PIcs;
    float gl1 = (ov[35][pr][pn] + ov[36][pr][pn] + ov[37][pr][pn]) * (1.f / 3.f);
    float val = 200.0f * (0.025f * ms + 0.975f * gl1);

    red[tid] = val;
    __syncthreads();
    for (int s = 128; s > 0; s >>= 1) {
        if (tid < s) red[tid] += red[tid + s];
        __syncthreads();
    }
    if (tid == 0)
        atomicAdd(out, red[0] * (1.0f / (8.0f * 512.0f * 512.0f)));
}

// ---------------------------------------------------------------------------
extern "C" void kernel_launch(void* const* d_in, const int* in_sizes, int n_in,
                              void* d_out, int out_size, void* d_ws, size_t ws_size,
                              hipStream_t stream) {
    const float* x  = (const float*)d_in[0];
    const float* y  = (const float*)d_in[1];
    const float* gb = (const float*)d_in[2];
    float* out = (float*)d_out;
    float* f1d = (float*)d_ws;                       // 5*33 floats of scratch

    (void)hipMemsetAsync(d_out, 0, sizeof(float), stream); // capture-safe
    extract_g1<<<dim3(1), dim3(32), 0, stream>>>(gb, f1d);
    msssim_l1_fused<<<dim3(BN * 32 * 32), dim3(256), 0, stream>>>(x, y, f1d, out);
}